// NN_G_88656714925147
// MI455X (gfx1250) — compile-verified
//
#include <hip/hip_runtime.h>
#include <hip/hip_bf16.h>

typedef __attribute__((ext_vector_type(2))) float v2f;
typedef __attribute__((ext_vector_type(8))) float v8f;

#define M_Q  1024   // queries
#define T_Q  64     // query time steps
#define F_Q  128    // features (GEMM K)
#define N_DB 4096   // database rows
#define TP   64     // pose time
#define PP   96     // pose feats

// ---------------------------------------------------------------------------
// q[m,f] = mean_t x[m,t,f]; float4 vectorized, coalesced over f.
// ---------------------------------------------------------------------------
__global__ void q_mean_kernel(const float* __restrict__ x, float* __restrict__ q) {
  int tid = blockIdx.x * blockDim.x + threadIdx.x;  // M*F/4 threads total
  int m = tid >> 5;                                 // F/4 = 32 float4 per row
  int c = tid & 31;
  const float4* xr = (const float4*)(x + (size_t)m * T_Q * F_Q) + c;
  float4 s = make_float4(0.f, 0.f, 0.f, 0.f);
#pragma unroll 8
  for (int t = 0; t < T_Q; ++t) {
    float4 v = xr[(size_t)t * (F_Q / 4)];
    s.x += v.x; s.y += v.y; s.z += v.z; s.w += v.w;
  }
  const float inv = 1.0f / (float)T_Q;
  s.x *= inv; s.y *= inv; s.z *= inv; s.w *= inv;
  ((float4*)q)[tid] = s;
}

// ---------------------------------------------------------------------------
// norms[n] = sum_f audio[n,f]^2 ; one wave32 per row, float4 per lane.
// ---------------------------------------------------------------------------
__global__ void audio_norm_kernel(const float* __restrict__ audio,
                                  float* __restrict__ norms) {
  int gid = blockIdx.x * blockDim.x + threadIdx.x;
  int n = gid >> 5;
  int lane = threadIdx.x & 31;
  float4 v = ((const float4*)(audio + (size_t)n * F_Q))[lane];
  float s = v.x * v.x + v.y * v.y + v.z * v.z + v.w * v.w;
#pragma unroll
  for (int off = 16; off > 0; off >>= 1) s += __shfl_xor(s, off, 32);
  if (lane == 0) norms[n] = s;
}

// Monotonic float -> uint mapping so u64 (key<<32|n) min == (min score, min n).
__device__ __forceinline__ unsigned orderedFloat(float f) {
  unsigned b = __float_as_uint(f);
  return b ^ (unsigned)(((int)b >> 31) | 0x80000000);
}

// ---------------------------------------------------------------------------
// Fused GEMM (audio @ q^T via V_WMMA_F32_16X16X4_F32) + argmin epilogue.
// One block = one 16-query M-tile; 8 waves stripe the 256 N-tiles.
// ---------------------------------------------------------------------------
__global__ void __launch_bounds__(256) argmin_wmma_kernel(
    const float* __restrict__ q, const float* __restrict__ audio,
    const float* __restrict__ norms, unsigned* __restrict__ idxOut) {
  __shared__ unsigned long long red[16][8];
  const int wave  = threadIdx.x >> 5;
  const int lane  = threadIdx.x & 31;
  const int lo    = lane & 15;   // row/col within 16
  const int hi    = lane >> 4;   // half-wave select (K pair / M+8 in C layout)
  const int mBase = blockIdx.x * 16;

  // A fragments (16x4 f32): lanes 0-15 hold M rows with K={k,k+1},
  // lanes 16-31 hold K={k+2,k+3}. Loop-invariant over N -> keep in VGPRs.
  v2f afrag[32];
  {
    const float* qr = q + (size_t)(mBase + lo) * F_Q + 2 * hi;
#pragma unroll
    for (int kk = 0; kk < 32; ++kk) afrag[kk] = *(const v2f*)(qr + 4 * kk);
  }

  unsigned long long best[8];
#pragma unroll
  for (int v = 0; v < 8; ++v) best[v] = ~0ull;

  for (int tile = wave; tile < N_DB / 16; tile += 8) {
    const int n = tile * 16 + lo;                 // this lane's DB row (B col)
    const float* ar = audio + (size_t)n * F_Q + 2 * hi;
    const float nrm = norms[n];
    v8f acc0 = {}, acc1 = {}, acc2 = {}, acc3 = {};   // 4 chains hide XDL RAW
#pragma unroll
    for (int kk = 0; kk < 32; kk += 4) {
      v2f b0 = *(const v2f*)(ar + 4 * (kk + 0));
      v2f b1 = *(const v2f*)(ar + 4 * (kk + 1));
      v2f b2 = *(const v2f*)(ar + 4 * (kk + 2));
      v2f b3 = *(const v2f*)(ar + 4 * (kk + 3));
      acc0 = __builtin_amdgcn_wmma_f32_16x16x4_f32(false, afrag[kk + 0], false, b0,
                                                   (short)0, acc0, false, false);
      acc1 = __builtin_amdgcn_wmma_f32_16x16x4_f32(false, afrag[kk + 1], false, b1,
                                                   (short)0, acc1, false, false);
      acc2 = __builtin_amdgcn_wmma_f32_16x16x4_f32(false, afrag[kk + 2], false, b2,
                                                   (short)0, acc2, false, false);
      acc3 = __builtin_amdgcn_wmma_f32_16x16x4_f32(false, afrag[kk + 3], false, b3,
                                                   (short)0, acc3, false, false);
    }
    v8f acc = (acc0 + acc1) + (acc2 + acc3);

    // D layout: VGPR v, lane<16 -> (M=v, N=lo); lane>=16 -> (M=v+8, N=lo).
#pragma unroll
    for (int v = 0; v < 8; ++v) {
      float s = nrm - 2.0f * acc[v];             // +||q||^2, /F are argmin-invariant
      unsigned long long key =
          ((unsigned long long)orderedFloat(s) << 32) | (unsigned)n;
      best[v] = key < best[v] ? key : best[v];
    }
  }

  // Reduce within each 16-lane half (xor masks 8..1 never cross halves).
#pragma unroll
  for (int v = 0; v < 8; ++v) {
#pragma unroll
    for (int off = 8; off > 0; off >>= 1) {
      unsigned long long o = __shfl_xor(best[v], off, 32);
      best[v] = o < best[v] ? o : best[v];
    }
  }
  if (lo == 0) {           // lanes 0 (M=v) and 16 (M=v+8)
#pragma unroll
    for (int v = 0; v < 8; ++v) red[v + 8 * hi][wave] = best[v];
  }
  __syncthreads();
  if (threadIdx.x < 16) {  // reduce across the 8 waves
    unsigned long long k = red[threadIdx.x][0];
#pragma unroll
    for (int w = 1; w < 8; ++w) {
      unsigned long long o = red[threadIdx.x][w];
      k = o < k ? o : k;
    }
    idxOut[mBase + threadIdx.x] = (unsigned)(k & 0xFFFFFFFFu);
  }
}

// ---------------------------------------------------------------------------
// out[m] = pose[idx[m]] (24 KB/row, float4 copies); last element = dummy[0].
// ---------------------------------------------------------------------------
__global__ void gather_kernel(const float* __restrict__ pose,
                              const unsigned* __restrict__ idx,
                              const float* __restrict__ dummy,
                              float* __restrict__ out, int out_size) {
  const int m = blockIdx.x;
  const unsigned row = idx[m];
  const float4* src = (const float4*)(pose + (size_t)row * TP * PP);
  float4* dst = (float4*)(out + (size_t)m * TP * PP);
  const int n4 = TP * PP / 4;  // 1536
  for (int i = threadIdx.x; i < n4; i += blockDim.x) dst[i] = src[i];
  if (m == 0 && threadIdx.x == 0) out[out_size - 1] = dummy[0];
}

extern "C" void kernel_launch(void* const* d_in, const int* in_sizes, int n_in,
                              void* d_out, int out_size, void* d_ws, size_t ws_size,
                              hipStream_t stream) {
  (void)in_sizes; (void)n_in; (void)ws_size;
  const float* x     = (const float*)d_in[0];
  const float* audio = (const float*)d_in[2];
  const float* pose  = (const float*)d_in[3];
  const float* dummy = (const float*)d_in[4];
  float* out = (float*)d_out;

  float*    q      = (float*)d_ws;               // M*F floats   (512 KB)
  float*    norms  = q + M_Q * F_Q;              // N floats     (16 KB)
  unsigned* idxbuf = (unsigned*)(norms + N_DB);  // M uints      (4 KB)

  q_mean_kernel<<<(M_Q * F_Q / 4) / 256, 256, 0, stream>>>(x, q);
  audio_norm_kernel<<<(N_DB * 32) / 256, 256, 0, stream>>>(audio, norms);
  argmin_wmma_kernel<<<M_Q / 16, 256, 0, stream>>>(q, audio, norms, idxbuf);
  gather_kernel<<<M_Q, 256, 0, stream>>>(pose, idxbuf, dummy, out, out_size);
}